// WeightedSVDHead_41145786696233
// MI455X (gfx1250) — compile-verified
//
#include <hip/hip_runtime.h>
#include <math.h>

// ---------------------------------------------------------------------------
// Weighted Kabsch / Procrustes head for MI455X (gfx1250, wave32).
// B=512 batches, N=16384 points. Memory-bound: ~235 MB streamed once
// (~10 us at 23.3 TB/s). Single-pass raw-moment reduction + tiny per-batch
// 3x3 SVD. Cross-thread reduction done with chained V_WMMA_F32_16X16X4_F32
// (all-ones A => column sums; K-row lane mapping of B is then irrelevant).
// ---------------------------------------------------------------------------

#define WSVD_EPS 1e-4f
#define WSVD_N   16384
#define WSVD_T   256     // threads per reduction block (8 wave32)

typedef __attribute__((ext_vector_type(2))) float v2f;
typedef __attribute__((ext_vector_type(8))) float v8f;

// ---------------------------------------------------------------------------
// Kernel 1: per-batch raw moments -> cov[9], src_mean[3], corres_mean[3]
// ws layout per batch: 15 floats.
// ---------------------------------------------------------------------------
__global__ __launch_bounds__(WSVD_T) void wsvd_reduce_kernel(
    const float* __restrict__ src,      // [B, N, 3]
    const float* __restrict__ corres,   // [B, N, 3]
    const float* __restrict__ weights,  // [B, N]
    float* __restrict__ ws)             // [B, 15]
{
    __shared__ float lds[WSVD_T][16];   // 16 KB partials

    const int b   = blockIdx.x;
    const int tid = threadIdx.x;

    // 16B-aligned bases: b*N*3*4 and b*N*4 bytes are multiples of 16.
    const float4* __restrict__ sp = (const float4*)(src    + (size_t)b * WSVD_N * 3);
    const float4* __restrict__ cp = (const float4*)(corres + (size_t)b * WSVD_N * 3);
    const float4* __restrict__ wp = (const float4*)(weights + (size_t)b * WSVD_N);

    float acc[16];
#pragma unroll
    for (int i = 0; i < 16; ++i) acc[i] = 0.0f;

    // 4 points per group: 3 float4 of src, 3 of corres, 1 float4 of weights.
    const int ngroups = WSVD_N / 4; // 4096
    for (int g = tid; g < ngroups; g += WSVD_T) {
        // prefetch next tile (speculative; silently dropped past the end)
        __builtin_prefetch(&sp[3 * (g + WSVD_T)], 0, 0);
        __builtin_prefetch(&cp[3 * (g + WSVD_T)], 0, 0);

        const float4 w4 = wp[g];
        const float4 s0 = sp[3 * g + 0], s1 = sp[3 * g + 1], s2 = sp[3 * g + 2];
        const float4 c0 = cp[3 * g + 0], c1 = cp[3 * g + 1], c2 = cp[3 * g + 2];

        const float psv[4][3] = {{s0.x, s0.y, s0.z}, {s0.w, s1.x, s1.y},
                                 {s1.z, s1.w, s2.x}, {s2.y, s2.z, s2.w}};
        const float pcv[4][3] = {{c0.x, c0.y, c0.z}, {c0.w, c1.x, c1.y},
                                 {c1.z, c1.w, c2.x}, {c2.y, c2.z, c2.w}};
        const float wv[4] = {w4.x, w4.y, w4.z, w4.w};

#pragma unroll
        for (int k = 0; k < 4; ++k) {
            const float w = wv[k];
            acc[0] += w;
            const float ws0 = w * psv[k][0];
            const float ws1 = w * psv[k][1];
            const float ws2 = w * psv[k][2];
            acc[1] += ws0; acc[2] += ws1; acc[3] += ws2;
            acc[4] += w * pcv[k][0];
            acc[5] += w * pcv[k][1];
            acc[6] += w * pcv[k][2];
            acc[7]  += ws0 * pcv[k][0]; acc[8]  += ws0 * pcv[k][1]; acc[9]  += ws0 * pcv[k][2];
            acc[10] += ws1 * pcv[k][0]; acc[11] += ws1 * pcv[k][1]; acc[12] += ws1 * pcv[k][2];
            acc[13] += ws2 * pcv[k][0]; acc[14] += ws2 * pcv[k][1]; acc[15] += ws2 * pcv[k][2];
        }
    }

#pragma unroll
    for (int i = 0; i < 16; ++i) lds[tid][i] = acc[i];
    __syncthreads();

    // Wave 0 (lanes 0..31, wave-uniform branch => EXEC all ones inside):
    // collapse 256x16 partials with chained V_WMMA_F32_16X16X4_F32.
    // A = all-ones 16x4, so D[m][n] = sum_k B[k][n] + C[m][n]: any lane<->K
    // permutation of the B operand gives the same result; only the column
    // mapping N = lane&15 (matches documented C/D layout) matters.
    if (tid < 32) {
        v2f a; a[0] = 1.0f; a[1] = 1.0f;
        v8f c = {};
        const int half = tid >> 4;   // which 16-lane half
        const int col  = tid & 15;   // accumulator index (N)
        const float* base = &lds[0][0];
#pragma unroll 4
        for (int g = 0; g < 64; ++g) {
            v2f bm;
            bm[0] = base[(4 * g + 0 + half) * 16 + col]; // rows 4g,   4g+1
            bm[1] = base[(4 * g + 2 + half) * 16 + col]; // rows 4g+2, 4g+3
            c = __builtin_amdgcn_wmma_f32_16x16x4_f32(
                    /*neg_a=*/false, a, /*neg_b=*/false, bm,
                    /*c_mod=*/(short)0, c, /*reuse_a=*/false, /*reuse_b=*/false);
        }
        // D row M=0 lives in c[0] on lanes 0..15 (N = lane).
        if (tid < 16) lds[0][tid] = c[0];
    }
    __syncthreads();

    if (tid == 0) {
        float t[16];
#pragma unroll
        for (int i = 0; i < 16; ++i) t[i] = lds[0][i];

        const float W0     = t[0];
        const float sum_w  = W0 + WSVD_EPS;        // matches reference
        const float inv_sw = 1.0f / sum_w;
        const float denom  = W0 * inv_sw + WSVD_EPS;
        const float inv_d  = 1.0f / denom;

        float sm[3], cm[3];
#pragma unroll
        for (int i = 0; i < 3; ++i) {
            sm[i] = t[1 + i] * inv_sw * inv_d;     // src_mean
            cm[i] = t[4 + i] * inv_sw * inv_d;     // corres_mean
        }

        float* o = ws + (size_t)b * 15;
        const float wnorm = W0 * inv_sw;           // sum of normalized w
#pragma unroll
        for (int i = 0; i < 3; ++i) {
#pragma unroll
            for (int j = 0; j < 3; ++j) {
                // cov = Sum w s c^T - sm (Sum w c)^T - (Sum w s) cm^T + (Sum w) sm cm^T
                const float cov = t[7 + 3 * i + j] * inv_sw
                                - sm[i] * (t[4 + j] * inv_sw)
                                - (t[1 + i] * inv_sw) * cm[j]
                                + wnorm * sm[i] * cm[j];
                o[3 * i + j] = cov;
            }
        }
#pragma unroll
        for (int i = 0; i < 3; ++i) { o[9 + i] = sm[i]; o[12 + i] = cm[i]; }
    }
}

// ---------------------------------------------------------------------------
// Kernel 2: one thread per batch -> 3x3 SVD (Jacobi on cov^T cov), rotation
// with reflection fix, translation.
// ---------------------------------------------------------------------------
__global__ void wsvd_solve_kernel(const float* __restrict__ ws,
                                  float* __restrict__ out, int B)
{
    const int b = blockIdx.x * blockDim.x + threadIdx.x;
    if (b >= B) return;

    const float* p = ws + (size_t)b * 15;
    float A[3][3];
#pragma unroll
    for (int i = 0; i < 3; ++i)
#pragma unroll
        for (int j = 0; j < 3; ++j) A[i][j] = p[3 * i + j];
    const float sm[3] = {p[9], p[10], p[11]};
    const float cm[3] = {p[12], p[13], p[14]};

    // S = A^T A (symmetric PSD)
    float S[3][3];
#pragma unroll
    for (int i = 0; i < 3; ++i)
#pragma unroll
        for (int j = 0; j < 3; ++j)
            S[i][j] = A[0][i] * A[0][j] + A[1][i] * A[1][j] + A[2][i] * A[2][j];

    float V[3][3] = {{1.f, 0.f, 0.f}, {0.f, 1.f, 0.f}, {0.f, 0.f, 1.f}};

    auto rot = [&](int pi, int qi) {
        const float apq = S[pi][qi];
        if (fabsf(apq) < 1e-30f) return;
        const float app = S[pi][pi], aqq = S[qi][qi];
        const float tau = (aqq - app) / (2.0f * apq);
        const float tt  = (tau >= 0.0f) ? 1.0f / (tau + sqrtf(1.0f + tau * tau))
                                        : 1.0f / (tau - sqrtf(1.0f + tau * tau));
        const float cth = 1.0f / sqrtf(1.0f + tt * tt);
        const float sth = tt * cth;
#pragma unroll
        for (int k = 0; k < 3; ++k) {            // S = S * J
            const float skp = S[k][pi], skq = S[k][qi];
            S[k][pi] = cth * skp - sth * skq;
            S[k][qi] = sth * skp + cth * skq;
        }
#pragma unroll
        for (int k = 0; k < 3; ++k) {            // S = J^T * S
            const float spk = S[pi][k], sqk = S[qi][k];
            S[pi][k] = cth * spk - sth * sqk;
            S[qi][k] = sth * spk + cth * sqk;
        }
#pragma unroll
        for (int k = 0; k < 3; ++k) {            // V = V * J
            const float vkp = V[k][pi], vkq = V[k][qi];
            V[k][pi] = cth * vkp - sth * vkq;
            V[k][qi] = sth * vkp + cth * vkq;
        }
    };

    for (int sweep = 0; sweep < 12; ++sweep) { rot(0, 1); rot(0, 2); rot(1, 2); }

    float lam[3] = {S[0][0], S[1][1], S[2][2]};
    // sort eigenpairs descending (swap V columns along)
#pragma unroll
    for (int i = 0; i < 2; ++i) {
        int m = i;
        for (int j = i + 1; j < 3; ++j) if (lam[j] > lam[m]) m = j;
        if (m != i) {
            const float tl = lam[i]; lam[i] = lam[m]; lam[m] = tl;
#pragma unroll
            for (int k = 0; k < 3; ++k) {
                const float tv = V[k][i]; V[k][i] = V[k][m]; V[k][m] = tv;
            }
        }
    }

    float sv[3], U[3][3];
#pragma unroll
    for (int j = 0; j < 3; ++j) {
        sv[j] = sqrtf(fmaxf(lam[j], 0.0f));
        const float inv = (sv[j] > 1e-20f) ? 1.0f / sv[j] : 0.0f;
#pragma unroll
        for (int i = 0; i < 3; ++i)
            U[i][j] = (A[i][0] * V[0][j] + A[i][1] * V[1][j] + A[i][2] * V[2][j]) * inv;
    }
    // degenerate third column -> cross product (r is invariant to its sign)
    if (sv[2] <= 1e-6f * sv[0] || sv[0] <= 1e-30f) {
        float cx = U[1][0] * U[2][1] - U[2][0] * U[1][1];
        float cy = U[2][0] * U[0][1] - U[0][0] * U[2][1];
        float cz = U[0][0] * U[1][1] - U[1][0] * U[0][1];
        const float nn = sqrtf(cx * cx + cy * cy + cz * cz);
        const float in = (nn > 1e-30f) ? 1.0f / nn : 0.0f;
        U[0][2] = cx * in; U[1][2] = cy * in; U[2][2] = cz * in;
    }

    auto det3 = [](const float M[3][3]) -> float {
        return M[0][0] * (M[1][1] * M[2][2] - M[1][2] * M[2][1])
             - M[0][1] * (M[1][0] * M[2][2] - M[1][2] * M[2][0])
             + M[0][2] * (M[1][0] * M[2][1] - M[1][1] * M[2][0]);
    };
    const float d = det3(U) * det3(V);   // = det(vh @ u^T)
    const float dg[3] = {1.0f, 1.0f, d};

    float R[3][3];
#pragma unroll
    for (int i = 0; i < 3; ++i)
#pragma unroll
        for (int j = 0; j < 3; ++j)
            R[i][j] = V[i][0] * dg[0] * U[j][0]
                    + V[i][1] * dg[1] * U[j][1]
                    + V[i][2] * dg[2] * U[j][2];

    float* ro = out + (size_t)b * 9;
#pragma unroll
    for (int i = 0; i < 3; ++i)
#pragma unroll
        for (int j = 0; j < 3; ++j) ro[3 * i + j] = R[i][j];

    float* to = out + (size_t)B * 9 + (size_t)b * 3;
#pragma unroll
    for (int i = 0; i < 3; ++i)
        to[i] = cm[i] - (R[i][0] * sm[0] + R[i][1] * sm[1] + R[i][2] * sm[2]);
}

// ---------------------------------------------------------------------------
extern "C" void kernel_launch(void* const* d_in, const int* in_sizes, int n_in,
                              void* d_out, int out_size, void* d_ws, size_t ws_size,
                              hipStream_t stream) {
    const float* src     = (const float*)d_in[0];
    const float* corres  = (const float*)d_in[1];
    const float* weights = (const float*)d_in[2];
    float* out = (float*)d_out;
    float* ws  = (float*)d_ws;

    const int B = in_sizes[2] / WSVD_N;   // 512

    wsvd_reduce_kernel<<<B, WSVD_T, 0, stream>>>(src, corres, weights, ws);
    wsvd_solve_kernel<<<(B + 63) / 64, 64, 0, stream>>>(ws, out, B);
}